// FusedEncoding_65833258713324
// MI455X (gfx1250) — compile-verified
//
#include <hip/hip_runtime.h>
#include <cmath>
#include <cstddef>

#define NLEVELS 16
#define LOG2T   19
#define TSIZE   (1u << LOG2T)
#define TMASK   (TSIZE - 1u)
#define PRIME1  2654435761u
#define PRIME2  805459861u
#define BLOCK   256

struct ResArr { float r[NLEVELS]; };

// Per-level corner hashing: shares the two v_mul_lo_u32 across all 8 corners.
__device__ __forceinline__ void corners(float ux, float uy, float uz, float r,
                                        unsigned idx[8],
                                        float& wx, float& wy, float& wz) {
    float px = ux * r, py = uy * r, pz = uz * r;
    float fx = floorf(px), fy = floorf(py), fz = floorf(pz);
    wx = px - fx; wy = py - fy; wz = pz - fz;
    unsigned x0 = (unsigned)(int)fx, x1 = x0 + 1u;
    unsigned hy0 = (unsigned)(int)fy * PRIME1, hy1 = hy0 + PRIME1;
    unsigned hz0 = (unsigned)(int)fz * PRIME2, hz1 = hz0 + PRIME2;
    // corner c = ox*4 + oy*2 + oz  (matches np.meshgrid 'ij' offsets)
    idx[0] = (x0 ^ hy0 ^ hz0) & TMASK;
    idx[1] = (x0 ^ hy0 ^ hz1) & TMASK;
    idx[2] = (x0 ^ hy1 ^ hz0) & TMASK;
    idx[3] = (x0 ^ hy1 ^ hz1) & TMASK;
    idx[4] = (x1 ^ hy0 ^ hz0) & TMASK;
    idx[5] = (x1 ^ hy0 ^ hz1) & TMASK;
    idx[6] = (x1 ^ hy1 ^ hz0) & TMASK;
    idx[7] = (x1 ^ hy1 ^ hz1) & TMASK;
}

__global__ void __launch_bounds__(BLOCK)
hashgrid_encode(const float* __restrict__ inp,
                const float* __restrict__ params,
                float* __restrict__ out,
                int npts, ResArr res) {
    __shared__ float sU[BLOCK * 3];   // 3 KB coordinate tile

    const int tid = threadIdx.x;
    const long long blockStart = (long long)blockIdx.x * BLOCK;
    const bool fullTile = (blockStart + BLOCK) <= (long long)npts;

    // --- CDNA5 async copy: stage this block's coords into LDS (ASYNCcnt path).
    // 192 lanes x 16B = 3072B; waves 6/7 have EXEC==0 and skip the VMEM op.
    if (fullTile) {
        if (tid < (BLOCK * 3) / 4) {
            unsigned ldsAddr = (unsigned)(size_t)(&sU[0]) + (unsigned)(tid * 16);
            const char* gaddr = (const char*)(inp + blockStart * 3) + (size_t)tid * 16;
            asm volatile("global_load_async_to_lds_b128 %[lds], %[g], off"
                         :: [lds] "v"(ldsAddr), [g] "v"(gaddr)
                         : "memory");
        }
        asm volatile("s_wait_asynccnt 0x0" ::: "memory");
    }
    __syncthreads();

    const long long n = blockStart + tid;
    if (n >= npts) return;

    float ux, uy, uz;
    if (fullTile) {
        // Explicit DS reads of the async-staged tile (the async write is invisible
        // to LLVM alias analysis, so read via asm; wait handled inside).
        // 12B/lane stride is bank-conflict-free: 3i is a bijection mod 64.
        unsigned caddr = (unsigned)(size_t)(&sU[0]) + (unsigned)(tid * 12);
        asm volatile("ds_load_b32 %0, %3\n\t"
                     "ds_load_b32 %1, %3 offset:4\n\t"
                     "ds_load_b32 %2, %3 offset:8\n\t"
                     "s_wait_dscnt 0x0"
                     : "=&v"(ux), "=&v"(uy), "=&v"(uz)
                     : "v"(caddr)
                     : "memory");
    } else {
        const float* p = inp + n * 3;
        ux = p[0]; uy = p[1]; uz = p[2];
    }
    ux = fminf(fmaxf(ux, 0.0f), 1.0f);
    uy = fminf(fmaxf(uy, 0.0f), 1.0f);
    uz = fminf(fmaxf(uz, 0.0f), 1.0f);

    const float2* __restrict__ tbl = (const float2*)params;

    // Depth-2 software pipeline: while level l's math runs, the 8 gathers for
    // BOTH l+1 and l+2 are in flight (loads retire in order -> the wait before
    // consuming slot l is loadcnt<=8 with l+1's already outstanding).
    float2  fbuf[2][8];
    float   wbuf[2][3];
    unsigned nidx[8];

    {
        float wx, wy, wz;
        corners(ux, uy, uz, res.r[0], nidx, wx, wy, wz);
        wbuf[0][0] = wx; wbuf[0][1] = wy; wbuf[0][2] = wz;
#pragma unroll
        for (int c = 0; c < 8; ++c) fbuf[0][c] = tbl[nidx[c]];

        corners(ux, uy, uz, res.r[1], nidx, wx, wy, wz);
        wbuf[1][0] = wx; wbuf[1][1] = wy; wbuf[1][2] = wz;
        const float2* t1 = tbl + (size_t)TSIZE;
#pragma unroll
        for (int c = 0; c < 8; ++c) fbuf[1][c] = t1[nidx[c]];
    }

    float acc[2 * NLEVELS];

#pragma unroll
    for (int l = 0; l < NLEVELS; ++l) {
        const int s = l & 1;
        const float wx = wbuf[s][0], wy = wbuf[s][1], wz = wbuf[s][2];
        const float2 f0 = fbuf[s][0], f1 = fbuf[s][1], f2 = fbuf[s][2], f3 = fbuf[s][3];
        const float2 f4 = fbuf[s][4], f5 = fbuf[s][5], f6 = fbuf[s][6], f7 = fbuf[s][7];

        if (l + 2 < NLEVELS) {  // refill this slot with level l+2's gathers
            float twx, twy, twz;
            corners(ux, uy, uz, res.r[l + 2], nidx, twx, twy, twz);
            wbuf[s][0] = twx; wbuf[s][1] = twy; wbuf[s][2] = twz;
            const float2* tn = tbl + (size_t)(l + 2) * TSIZE;
#pragma unroll
            for (int c = 0; c < 8; ++c) fbuf[s][c] = tn[nidx[c]];
        }

        // trilinear weights (product structure: 12 muls for 8 weights)
        const float wx0 = 1.0f - wx, wy0 = 1.0f - wy, wz0 = 1.0f - wz;
        const float a00 = wy0 * wz0, a01 = wy0 * wz, a10 = wy * wz0, a11 = wy * wz;
        const float w0 = wx0 * a00, w1 = wx0 * a01, w2 = wx0 * a10, w3 = wx0 * a11;
        const float w4 = wx  * a00, w5 = wx  * a01, w6 = wx  * a10, w7 = wx  * a11;

        float o0 = f0.x * w0, o1 = f0.y * w0;
        o0 = fmaf(f1.x, w1, o0); o1 = fmaf(f1.y, w1, o1);
        o0 = fmaf(f2.x, w2, o0); o1 = fmaf(f2.y, w2, o1);
        o0 = fmaf(f3.x, w3, o0); o1 = fmaf(f3.y, w3, o1);
        o0 = fmaf(f4.x, w4, o0); o1 = fmaf(f4.y, w4, o1);
        o0 = fmaf(f5.x, w5, o0); o1 = fmaf(f5.y, w5, o1);
        o0 = fmaf(f6.x, w6, o0); o1 = fmaf(f6.y, w6, o1);
        o0 = fmaf(f7.x, w7, o0); o1 = fmaf(f7.y, w7, o1);

        acc[2 * l + 0] = o0;
        acc[2 * l + 1] = o1;
    }

    // 8 x b128 stores; each row is 128B-aligned and contiguous.
    float4* op = (float4*)(out + (size_t)n * (2 * NLEVELS));
#pragma unroll
    for (int j = 0; j < 8; ++j)
        op[j] = make_float4(acc[4 * j + 0], acc[4 * j + 1],
                            acc[4 * j + 2], acc[4 * j + 3]);
}

extern "C" void kernel_launch(void* const* d_in, const int* in_sizes, int n_in,
                              void* d_out, int out_size, void* d_ws, size_t ws_size,
                              hipStream_t stream) {
    const float* inp    = (const float*)d_in[0];
    const float* params = (const float*)d_in[1];
    float* out          = (float*)d_out;
    const int npts = in_sizes[0] / 3;

    // Reproduce numpy's float64 resolution chain exactly on the host:
    // b = exp((log(512)-log(16))/15); res_l = floor(16 * b**l)
    ResArr ra;
    const double b = exp((log(512.0) - log(16.0)) / 15.0);
    for (int l = 0; l < NLEVELS; ++l)
        ra.r[l] = (float)floor(16.0 * pow(b, (double)l));

    const int grid = (npts + BLOCK - 1) / BLOCK;
    hashgrid_encode<<<grid, BLOCK, 0, stream>>>(inp, params, out, npts, ra);
}